// Net_23098334118132
// MI455X (gfx1250) — compile-verified
//
#include <hip/hip_runtime.h>
#include <math.h>

// Problem constants (from reference)
#define BGR   1024          // graphs
#define NPER  256           // nodes per graph
#define DEG   16
#define EDG_G (NPER*DEG)    // 4096 edges per graph (contiguous in edge_index)
#define NTOT  (BGR*NPER)    // 262144
#define ETOT  (NTOT*DEG)    // 4194304
#define FIN   7
#define HDIM  64
#define KCL   50
#define SST   52            // stored stride for S / AS in workspace

typedef __attribute__((ext_vector_type(2))) float v2f;
typedef __attribute__((ext_vector_type(8))) float v8f;

// ---------------- WMMA helpers: D = A(16xK) x B(Kx16) + C, fp32, K-step 4 ---
// A 16x4 fragment: lanes 0-15 -> M=lane (K = k0+0 / k0+1), lanes 16-31 -> M=lane-16 (K = k0+2 / k0+3)
// B 4x16 fragment mirrors A with N across lanes.  C/D: c[v] = (M = m0+v+8*(lane>=16), N = n0+(lane&15)).
__device__ __forceinline__ v8f wmma_ab(const float* A, int lda, int m0, int k0,
                                       const float* B, int ldb, int n0, v8f c) {
  const int lane = threadIdx.x & 31;
  const int l = lane & 15;
  const int kk = (lane >> 4) << 1;
  v2f a, b;
  a.x = A[(m0 + l) * lda + (k0 + kk)];
  a.y = A[(m0 + l) * lda + (k0 + kk + 1)];
  b.x = B[(k0 + kk) * ldb + (n0 + l)];
  b.y = B[(k0 + kk + 1) * ldb + (n0 + l)];
  return __builtin_amdgcn_wmma_f32_16x16x4_f32(false, a, false, b, (short)0, c, false, false);
}

// A supplied transposed: Alogical[m][k] = S[k*lds_ + m]   (used for S^T @ X GEMMs)
__device__ __forceinline__ v8f wmma_atb(const float* S, int lds_, int m0, int k0,
                                        const float* B, int ldb, int n0, v8f c) {
  const int lane = threadIdx.x & 31;
  const int l = lane & 15;
  const int kk = (lane >> 4) << 1;
  v2f a, b;
  a.x = S[(k0 + kk) * lds_ + (m0 + l)];
  a.y = S[(k0 + kk + 1) * lds_ + (m0 + l)];
  b.x = B[(k0 + kk) * ldb + (n0 + l)];
  b.y = B[(k0 + kk + 1) * ldb + (n0 + l)];
  return __builtin_amdgcn_wmma_f32_16x16x4_f32(false, a, false, b, (short)0, c, false, false);
}

__device__ __forceinline__ void wmma_st(float* D, int ldd, int m0, int n0, v8f c) {
  const int lane = threadIdx.x & 31;
  const int row = m0 + ((lane >> 4) << 3);
  const int col = n0 + (lane & 15);
#pragma unroll
  for (int v = 0; v < 8; ++v) D[(row + v) * ldd + col] = c[v];
}

// =============== K1: GIN scatter-add + 2-layer MLP -> h [N,64] ===============
__global__ void __launch_bounds__(256)
k1_gin(const float* __restrict__ x, const float* __restrict__ W1a,
       const float* __restrict__ b1a, const float* __restrict__ W1b,
       const float* __restrict__ b1b, const int* __restrict__ src,
       const int* __restrict__ dst, float* __restrict__ h_out) {
  __shared__ float xo[NPER * 8];        // original x (padded 7->8)
  __shared__ float xz[NPER * 8];        // x + aggregated neighbors
  __shared__ float T[NPER * 65];        // relu(layer1), stride 65 (A-side)
  __shared__ float Wa[8 * 64];
  __shared__ float Wb[64 * 64];
  __shared__ float ba[64], bb[64];
  const int g = blockIdx.x, tid = threadIdx.x;

  const float* xg = x + (size_t)g * NPER * FIN;
#pragma unroll
  for (int f = 0; f < FIN; ++f) { float v = xg[tid * FIN + f]; xo[tid * 8 + f] = v; xz[tid * 8 + f] = v; }
  xo[tid * 8 + 7] = 0.f; xz[tid * 8 + 7] = 0.f;
  for (int i = tid; i < FIN * 64; i += 256) Wa[i] = W1a[i];
  if (tid < 64) { Wa[FIN * 64 + tid] = 0.f; ba[tid] = b1a[tid]; bb[tid] = b1b[tid]; }
  for (int i = tid; i < 64 * 64; i += 256) Wb[i] = W1b[i];
  __syncthreads();

  // scatter-add: agg[dst] += x[src]   (all edges of graph g are contiguous)
  const int base = g * EDG_G;
  for (int e = tid; e < EDG_G; e += 256) {
    const int si = src[base + e] & (NPER - 1);
    const int di = dst[base + e] & (NPER - 1);
#pragma unroll
    for (int f = 0; f < FIN; ++f) atomicAdd(&xz[di * 8 + f], xo[si * 8 + f]);
  }
  __syncthreads();

  // layer 1 (K=7, VALU): T = relu(xz @ W1a + b1a)
  for (int j = 0; j < 64; ++j) {
    float acc = ba[j];
#pragma unroll
    for (int f = 0; f < FIN; ++f) acc += xz[tid * 8 + f] * Wa[f * 64 + j];
    T[tid * 65 + j] = fmaxf(acc, 0.f);
  }
  __syncthreads();

  // layer 2 (WMMA): h = T @ W1b + b1b   (M=256,N=64,K=64) -> global
  const int w = tid >> 5, lane = tid & 31;
  float* hg = h_out + (size_t)g * NPER * HDIM;
  for (int t = 0; t < 8; ++t) {
    const int tile = w * 8 + t, mt = tile >> 2, nt = tile & 3;
    v8f c = {};
    for (int k0 = 0; k0 < 64; k0 += 4) c = wmma_ab(T, 65, mt * 16, k0, Wb, 64, nt * 16, c);
    const int row = mt * 16 + ((lane >> 4) << 3);
    const int col = nt * 16 + (lane & 15);
#pragma unroll
    for (int v = 0; v < 8; ++v) hg[(row + v) * HDIM + col] = c[v] + bb[col];
  }
}

// =============== K2: pooling scores + softmax -> S [N,52], entropy partials ==
__global__ void __launch_bounds__(256)
k2_pool(const float* __restrict__ h_in, const float* __restrict__ Wp,
        const float* __restrict__ bp, float* __restrict__ S_out,
        float* __restrict__ entp) {
  __shared__ float Hs[NPER * 65];
  __shared__ float Wps[64 * 64];       // Wp zero-padded 50 -> 64 cols
  __shared__ float Ps[NPER * 65];
  __shared__ float bps[64];
  __shared__ float red[256];
  const int g = blockIdx.x, tid = threadIdx.x;

  const float* hg = h_in + (size_t)g * NPER * HDIM;
  for (int i = tid; i < NPER * 64; i += 256) Hs[(i >> 6) * 65 + (i & 63)] = hg[i];
  for (int i = tid; i < 64 * 64; i += 256) {
    const int r = i >> 6, c = i & 63;
    Wps[i] = (c < KCL) ? Wp[r * KCL + c] : 0.f;
  }
  if (tid < 64) bps[tid] = (tid < KCL) ? bp[tid] : 0.f;
  __syncthreads();

  const int w = tid >> 5, lane = tid & 31;
  for (int t = 0; t < 8; ++t) {
    const int tile = w * 8 + t, mt = tile >> 2, nt = tile & 3;
    v8f c = {};
    for (int k0 = 0; k0 < 64; k0 += 4) c = wmma_ab(Hs, 65, mt * 16, k0, Wps, 64, nt * 16, c);
    const int row = mt * 16 + ((lane >> 4) << 3);
    const int col = nt * 16 + (lane & 15);
#pragma unroll
    for (int v = 0; v < 8; ++v) Ps[(row + v) * 65 + col] = c[v] + bps[col];
  }
  __syncthreads();

  // per-row softmax over 50 valid clusters + entropy contribution
  float* Sg = S_out + (size_t)g * NPER * SST;
  float m = -1e30f;
  for (int k = 0; k < KCL; ++k) m = fmaxf(m, Ps[tid * 65 + k]);
  float s = 0.f;
  for (int k = 0; k < KCL; ++k) { float e = expf(Ps[tid * 65 + k] - m); Ps[tid * 65 + k] = e; s += e; }
  const float inv = 1.f / s;
  float ent = 0.f;
  for (int k = 0; k < KCL; ++k) {
    const float p = Ps[tid * 65 + k] * inv;
    Sg[tid * SST + k] = p;
    ent += p * logf(p + 1e-15f);
  }
  Sg[tid * SST + 50] = 0.f; Sg[tid * SST + 51] = 0.f;
  red[tid] = ent; __syncthreads();
  for (int o = 128; o > 0; o >>= 1) { if (tid < o) red[tid] += red[tid + o]; __syncthreads(); }
  if (tid == 0) entp[g] = red[0];
}

// =============== K3: sparse AS = A@S, ΣA², Σ_e <S[s],S[d]> =================
__global__ void __launch_bounds__(256)
k3_edges(const float* __restrict__ S_in, const int* __restrict__ src,
         const int* __restrict__ dst, float* __restrict__ AS_out,
         float* __restrict__ linkA) {
  __shared__ float smem3[2 * NPER * SST];   // [Ss | ASs], later aliased as u32 counts
  __shared__ float red[256];
  float* Ss = smem3;
  float* ASs = smem3 + NPER * SST;
  const int g = blockIdx.x, tid = threadIdx.x;

  const float* Sg = S_in + (size_t)g * NPER * SST;
  for (int i = tid; i < NPER * SST; i += 256) { Ss[i] = Sg[i]; ASs[i] = 0.f; }
  __syncthreads();

  const int base = g * EDG_G;
  int se[16], de[16];
#pragma unroll
  for (int j = 0; j < 16; ++j) {
    const int e = base + tid * 16 + j;
    se[j] = src[e] & (NPER - 1);
    de[j] = dst[e] & (NPER - 1);
  }
  float adot = 0.f;
  for (int j = 0; j < 16; ++j) {
    const float* srow = &Ss[se[j] * SST];
    const float* drow = &Ss[de[j] * SST];
    float* arow = &ASs[se[j] * SST];
    for (int k = 0; k < KCL; ++k) {
      const float dv = drow[k];
      atomicAdd(&arow[k], dv);               // AS[s] += S[d]  (A[s,d] += 1)
      adot += srow[k] * dv;                  // Σ_e SSt[s,d]
    }
  }
  __syncthreads();
  float* ASg = AS_out + (size_t)g * NPER * SST;
  for (int i = tid; i < NPER * SST; i += 256) ASg[i] = ASs[i];
  __syncthreads();

  // ΣA² via 64-row count chunks (alias counts over Ss/ASs region)
  unsigned int* cnt = (unsigned int*)smem3;  // 64*256 u32 = 64KB < 2*NPER*SST*4
  float a2 = 0.f;
  for (int ch = 0; ch < 4; ++ch) {
    for (int i = tid; i < 64 * 256; i += 256) cnt[i] = 0u;
    __syncthreads();
#pragma unroll
    for (int j = 0; j < 16; ++j)
      if ((se[j] >> 6) == ch) atomicAdd(&cnt[(se[j] & 63) * 256 + de[j]], 1u);
    __syncthreads();
    for (int i = tid; i < 64 * 256; i += 256) { const float v = (float)cnt[i]; a2 += v * v; }
    __syncthreads();
  }
  red[tid] = a2 - 2.f * adot; __syncthreads();
  for (int o = 128; o > 0; o >>= 1) { if (tid < o) red[tid] += red[tid + o]; __syncthreads(); }
  if (tid == 0) linkA[g] = red[0];           // + ||S^T S||_F² added in K4
}

// =============== K4: xp=SᵀH, Ap=SᵀAS, G-term, dense-GIN MLP, classifier =====
__global__ void __launch_bounds__(256)
k4_cluster(const float* __restrict__ h_in, const float* __restrict__ S_in,
           const float* __restrict__ AS_in, const float* __restrict__ W2a,
           const float* __restrict__ b2a, const float* __restrict__ W2b,
           const float* __restrict__ b2b, const float* __restrict__ Wl,
           const float* __restrict__ bl, float* __restrict__ linkA,
           float* __restrict__ out_logp) {
  __shared__ float sm4[28864];
  __shared__ float red[256];
  __shared__ float b2as[64], b2bs[64];
  // phase A view: Ss = sm4[0..16384)                (256 x 64, clusters padded)
  // phase B view: W2as|W2bs|h2|u2|o2 over sm4[0..20608)
  float* Ss   = sm4;
  float* W2as = sm4;
  float* W2bs = sm4 + 4096;
  float* h2   = sm4 + 8192;    // 64 x 65 (A-side)
  float* u2   = sm4 + 12352;   // 64 x 65 (A-side)
  float* o2   = sm4 + 16512;   // 64 x 64
  float* xp   = sm4 + 20608;   // 64 x 64
  float* Ap   = sm4 + 24704;   // 64 x 65 (A-side)
  const int g = blockIdx.x, tid = threadIdx.x, w = tid >> 5, lane = tid & 31;

  const float* Sg  = S_in  + (size_t)g * NPER * SST;
  const float* hg  = h_in  + (size_t)g * NPER * HDIM;   // B-side, read from L2
  const float* ASg = AS_in + (size_t)g * NPER * SST;    // B-side, read from L2
  for (int i = tid; i < NPER * 64; i += 256) {
    const int r = i >> 6, c = i & 63;
    Ss[i] = (c < KCL) ? Sg[r * SST + c] : 0.f;
  }
  if (tid < 64) { b2as[tid] = b2a[tid]; b2bs[tid] = b2b[tid]; }
  __syncthreads();

  // xp = Sᵀ @ H  (M=64,N=64,K=256); G-term = ||SᵀS||_F² accumulated on the fly
  float g2 = 0.f;
  for (int t = 0; t < 2; ++t) {
    const int tile = w * 2 + t, mt = tile >> 2, nt = tile & 3;
    v8f c = {};
    for (int k0 = 0; k0 < NPER; k0 += 4) c = wmma_atb(Ss, 64, mt * 16, k0, hg, HDIM, nt * 16, c);
    wmma_st(xp, 64, mt * 16, nt * 16, c);
    v8f cg = {};
    for (int k0 = 0; k0 < NPER; k0 += 4) cg = wmma_atb(Ss, 64, mt * 16, k0, Ss, 64, nt * 16, cg);
#pragma unroll
    for (int v = 0; v < 8; ++v) g2 += cg[v] * cg[v];
  }
  // Ap = Sᵀ @ AS ; AS cols beyond stored stride only ever multiply zero rows of xp
  for (int t = 0; t < 2; ++t) {
    const int tile = w * 2 + t, mt = tile >> 2, nt = tile & 3;
    v8f c = {};
    for (int k0 = 0; k0 < NPER; k0 += 4) {
      // guard B reads to the 52 stored columns; cols >= 50 of Ap multiply zero xp rows
      const int lane2 = threadIdx.x & 31;
      const int l = lane2 & 15, kk = (lane2 >> 4) << 1;
      v2f a, b;
      a.x = Ss[(k0 + kk) * 64 + mt * 16 + l];
      a.y = Ss[(k0 + kk + 1) * 64 + mt * 16 + l];
      const int nc = nt * 16 + l;
      b.x = (nc < SST) ? ASg[(k0 + kk) * SST + nc] : 0.f;
      b.y = (nc < SST) ? ASg[(k0 + kk + 1) * SST + nc] : 0.f;
      c = __builtin_amdgcn_wmma_f32_16x16x4_f32(false, a, false, b, (short)0, c, false, false);
    }
    wmma_st(Ap, 65, mt * 16, nt * 16, c);
  }
  // reduce the Gram term into linkA (barriers also fence Ss before overwrite)
  red[tid] = g2; __syncthreads();
  for (int o = 128; o > 0; o >>= 1) { if (tid < o) red[tid] += red[tid + o]; __syncthreads(); }
  if (tid == 0) linkA[g] += red[0];
  __syncthreads();

  // phase B: overwrite Ss region with weights
  for (int i = tid; i < 64 * 64; i += 256) { W2as[i] = W2a[i]; W2bs[i] = W2b[i]; }
  __syncthreads();

  // h2 = xp + Ap @ xp   (M=N=K=64)
  for (int t = 0; t < 2; ++t) {
    const int tile = w * 2 + t, mt = tile >> 2, nt = tile & 3;
    v8f c = {};
    for (int k0 = 0; k0 < 64; k0 += 4) c = wmma_ab(Ap, 65, mt * 16, k0, xp, 64, nt * 16, c);
    const int row = mt * 16 + ((lane >> 4) << 3);
    const int col = nt * 16 + (lane & 15);
#pragma unroll
    for (int v = 0; v < 8; ++v) h2[(row + v) * 65 + col] = c[v] + xp[(row + v) * 64 + col];
  }
  __syncthreads();
  // u2 = relu(h2 @ W2a + b2a)
  for (int t = 0; t < 2; ++t) {
    const int tile = w * 2 + t, mt = tile >> 2, nt = tile & 3;
    v8f c = {};
    for (int k0 = 0; k0 < 64; k0 += 4) c = wmma_ab(h2, 65, mt * 16, k0, W2as, 64, nt * 16, c);
    const int row = mt * 16 + ((lane >> 4) << 3);
    const int col = nt * 16 + (lane & 15);
#pragma unroll
    for (int v = 0; v < 8; ++v) u2[(row + v) * 65 + col] = fmaxf(c[v] + b2as[col], 0.f);
  }
  __syncthreads();
  // o2 = u2 @ W2b + b2b
  for (int t = 0; t < 2; ++t) {
    const int tile = w * 2 + t, mt = tile >> 2, nt = tile & 3;
    v8f c = {};
    for (int k0 = 0; k0 < 64; k0 += 4) c = wmma_ab(u2, 65, mt * 16, k0, W2bs, 64, nt * 16, c);
    const int row = mt * 16 + ((lane >> 4) << 3);
    const int col = nt * 16 + (lane & 15);
#pragma unroll
    for (int v = 0; v < 8; ++v) o2[(row + v) * 64 + col] = c[v] + b2bs[col];
  }
  __syncthreads();

  // mean over the 50 real clusters, classifier, log_softmax
  if (tid < 64) {
    float m = 0.f;
    for (int k = 0; k < KCL; ++k) m += o2[k * 64 + tid];
    red[tid] = m * (1.f / (float)KCL);
  }
  __syncthreads();
  if (tid == 0) {
    float l0 = bl[0], l1 = bl[1];
    for (int d = 0; d < 64; ++d) { l0 += red[d] * Wl[d * 2 + 0]; l1 += red[d] * Wl[d * 2 + 1]; }
    const float mx = fmaxf(l0, l1);
    const float lse = mx + logf(expf(l0 - mx) + expf(l1 - mx));
    out_logp[g * 2 + 0] = l0 - lse;
    out_logp[g * 2 + 1] = l1 - lse;
  }
}

// =============== K5: reduce aux loss ========================================
__global__ void __launch_bounds__(256)
k5_final(const float* __restrict__ linkA, const float* __restrict__ entp,
         float* __restrict__ out_aux) {
  __shared__ float r1[256], r2[256];
  const int tid = threadIdx.x;
  float a = 0.f, b = 0.f;
  for (int i = tid; i < BGR; i += 256) { a += linkA[i]; b += entp[i]; }
  r1[tid] = a; r2[tid] = b; __syncthreads();
  for (int o = 128; o > 0; o >>= 1) {
    if (tid < o) { r1[tid] += r1[tid + o]; r2[tid] += r2[tid + o]; }
    __syncthreads();
  }
  if (tid == 0) {
    const float link = sqrtf(fmaxf(r1[0], 0.f)) / (float)((size_t)BGR * NPER * NPER);
    const float ent = -r2[0] / (float)NTOT;
    out_aux[0] = link + ent;
  }
}

// ============================================================================
extern "C" void kernel_launch(void* const* d_in, const int* in_sizes, int n_in,
                              void* d_out, int out_size, void* d_ws, size_t ws_size,
                              hipStream_t stream) {
  (void)in_sizes; (void)n_in; (void)out_size; (void)ws_size;
  const float* x   = (const float*)d_in[0];
  const float* W1a = (const float*)d_in[1];
  const float* b1a = (const float*)d_in[2];
  const float* W1b = (const float*)d_in[3];
  const float* b1b = (const float*)d_in[4];
  const float* Wp  = (const float*)d_in[5];
  const float* bp  = (const float*)d_in[6];
  const float* W2a = (const float*)d_in[7];
  const float* b2a = (const float*)d_in[8];
  const float* W2b = (const float*)d_in[9];
  const float* b2b = (const float*)d_in[10];
  const float* Wl  = (const float*)d_in[11];
  const float* bl  = (const float*)d_in[12];
  const int*   ei  = (const int*)d_in[13];   // [2, E]
  const int*   src = ei;
  const int*   dst = ei + ETOT;

  float* h_ws  = (float*)d_ws;                               // N*64
  float* S_ws  = h_ws  + (size_t)NTOT * HDIM;                // N*52
  float* AS_ws = S_ws  + (size_t)NTOT * SST;                 // N*52
  float* linkA = AS_ws + (size_t)NTOT * SST;                 // B
  float* entp  = linkA + BGR;                                // B
  float* out   = (float*)d_out;                              // [2048] logp + [1] aux

  k1_gin    <<<BGR, 256, 0, stream>>>(x, W1a, b1a, W1b, b1b, src, dst, h_ws);
  k2_pool   <<<BGR, 256, 0, stream>>>(h_ws, Wp, bp, S_ws, entp);
  k3_edges  <<<BGR, 256, 0, stream>>>(S_ws, src, dst, AS_ws, linkA);
  k4_cluster<<<BGR, 256, 0, stream>>>(h_ws, S_ws, AS_ws, W2a, b2a, W2b, b2b,
                                      Wl, bl, linkA, out);
  k5_final  <<<1, 256, 0, stream>>>(linkA, entp, out + 2 * BGR);
}